// AveragedNormals_19567871001218
// MI455X (gfx1250) — compile-verified
//
#include <hip/hip_runtime.h>

typedef __attribute__((ext_vector_type(2))) float v2f;
typedef __attribute__((ext_vector_type(8))) float v8f;

#define NPTS 8192
#define KNN  128
#define QPB  4          // queries per block in the KNN/LRF kernel
#define NSAMP 2

// ---------------------------------------------------------------------------
// Kernel 0: pack (x,y,z,|p|^2) for WMMA operands
// ---------------------------------------------------------------------------
__global__ __launch_bounds__(256) void pack_pts_kernel(const float* __restrict__ v,
                                                       float4* __restrict__ pts4,
                                                       int total) {
  int g = blockIdx.x * blockDim.x + threadIdx.x;
  if (g < total) {
    float x = v[3 * g + 0], y = v[3 * g + 1], z = v[3 * g + 2];
    pts4[g] = make_float4(x, y, z, x * x + y * y + z * z);
  }
}

// ---------------------------------------------------------------------------
// Closed-form smallest eigenvector of a symmetric 3x3 matrix
// ---------------------------------------------------------------------------
__device__ __forceinline__ void smallest_evec(float a00, float a01, float a02,
                                              float a11, float a12, float a22,
                                              float* out) {
  float p1 = a01 * a01 + a02 * a02 + a12 * a12;
  float q  = (a00 + a11 + a22) * (1.0f / 3.0f);
  float b00 = a00 - q, b11 = a11 - q, b22 = a22 - q;
  float p2 = b00 * b00 + b11 * b11 + b22 * b22 + 2.0f * p1;
  float p  = sqrtf(fmaxf(p2 * (1.0f / 6.0f), 1e-30f));
  float inv = 1.0f / p;
  float c00 = b00 * inv, c01 = a01 * inv, c02 = a02 * inv;
  float c11 = b11 * inv, c12 = a12 * inv, c22 = b22 * inv;
  float detB = c00 * (c11 * c22 - c12 * c12)
             - c01 * (c01 * c22 - c12 * c02)
             + c02 * (c01 * c12 - c11 * c02);
  float r = fminf(1.0f, fmaxf(-1.0f, 0.5f * detB));
  float phi = acosf(r) * (1.0f / 3.0f);
  float e3 = q + 2.0f * p * cosf(phi + 2.0943951023931953f); // smallest eigenvalue
  // eigenvector = best cross product of rows of (A - e3*I)
  float m00 = a00 - e3, m11 = a11 - e3, m22 = a22 - e3;
  float v0x = a01 * a12 - a02 * m11, v0y = a02 * a01 - m00 * a12, v0z = m00 * m11 - a01 * a01;
  float v1x = a01 * m22 - a02 * a12, v1y = a02 * a02 - m00 * m22, v1z = m00 * a12 - a01 * a02;
  float v2x = m11 * m22 - a12 * a12, v2y = a12 * a02 - a01 * m22, v2z = a01 * a12 - m11 * a02;
  float n0 = v0x * v0x + v0y * v0y + v0z * v0z;
  float n1 = v1x * v1x + v1y * v1y + v1z * v1z;
  float n2 = v2x * v2x + v2y * v2y + v2z * v2z;
  float vx = v0x, vy = v0y, vz = v0z, nn = n0;
  if (n1 > nn) { vx = v1x; vy = v1y; vz = v1z; nn = n1; }
  if (n2 > nn) { vx = v2x; vy = v2y; vz = v2z; nn = n2; }
  if (nn < 1e-30f) { vx = 1.0f; vy = 0.0f; vz = 0.0f; nn = 1.0f; }
  float s = rsqrtf(nn);
  out[0] = vx * s; out[1] = vy * s; out[2] = vz * s;
}

// ---------------------------------------------------------------------------
// Kernel 1: WMMA distance tiles -> exact top-128 selection -> LRF normal (z)
//   block = 256 threads (8 wave32), handles QPB=4 query points
// ---------------------------------------------------------------------------
__global__ __launch_bounds__(256) void knn_lrf_kernel(const float4* __restrict__ pts4,
                                                      float4* __restrict__ normals,
                                                      unsigned short* __restrict__ idxOut) {
  __shared__ float d2s[QPB * NPTS];          // 128 KB of the 320 KB WGP LDS
  __shared__ int   sel_idx[QPB][KNN];
  __shared__ float sel_d2[QPB][KNN];
  __shared__ float4 qpt[QPB];
  __shared__ unsigned scnt[QPB];
  __shared__ int   gfill[QPB];
  __shared__ float covS[QPB][8];
  __shared__ float zv[QPB][3];
  __shared__ int   pcnt[QPB];
  __shared__ float radS[QPB];

  const int tid  = threadIdx.x;
  const int lane = tid & 31;
  const int wave = tid >> 5;
  const int half = lane >> 4;
  const int lm   = lane & 15;

  const int b     = blockIdx.x / (NPTS / QPB);
  const int qbase = (blockIdx.x % (NPTS / QPB)) * QPB;
  const float4* P = pts4 + (size_t)b * NPTS;

  if (tid < QPB) qpt[tid] = P[qbase + tid];
  __syncthreads();

  // ---- A operand (16x4 f32, per ISA lane layout): row m = (-2x,-2y,-2z, 1) ----
  v2f a; a.x = 0.0f; a.y = 0.0f;
  if (lm < QPB) {
    float4 qv = qpt[lm];
    if (half == 0) { a.x = -2.0f * qv.x; a.y = -2.0f * qv.y; }
    else           { a.x = -2.0f * qv.z; a.y = 1.0f; }
  }
  // ---- C init (16x16 f32 layout): row m preloaded with |q_m|^2 ----
  v8f cinit;
#pragma unroll
  for (int r = 0; r < 8; ++r) cinit[r] = 0.0f;
  if (half == 0) {
#pragma unroll
    for (int r = 0; r < QPB; ++r) cinit[r] = qpt[r].w;
  }

  // ---- distance tiles: one v_wmma_f32_16x16x4_f32 per 16 candidates ----
  for (int t = wave; t < NPTS / 16; t += 8) {
    int j0 = t * 16;
    float4 p = P[j0 + lm];                  // B col n = (x,y,z,|c|^2)
    v2f bb;
    if (half == 0) { bb.x = p.x; bb.y = p.y; }
    else           { bb.x = p.z; bb.y = p.w; }
    v8f d = __builtin_amdgcn_wmma_f32_16x16x4_f32(
        false, a, false, bb, (short)0, cinit, false, false);
    if (half == 0) {
#pragma unroll
      for (int r = 0; r < QPB; ++r)
        d2s[r * NPTS + j0 + lm] = fmaxf(d[r], 1e-12f);   // matches max(d2, EPS)
    }
  }
  __syncthreads();

  // ---- exact 128th order statistic via binary search on f32 bit pattern ----
  const int g = tid >> 6;   // query 0..3 (64 threads each)
  const int l = tid & 63;

  unsigned lo = 0u, hi = 0x7f7fffffu;
  for (int it = 0; it < 31; ++it) {
    unsigned mid = (lo + hi) >> 1;
    if (l == 0) scnt[g] = 0u;
    __syncthreads();
    unsigned c = 0u;
    for (int j = l; j < NPTS; j += 64)
      c += (__float_as_uint(d2s[g * NPTS + j]) <= mid) ? 1u : 0u;
    atomicAdd(&scnt[g], c);
    __syncthreads();
    if (scnt[g] >= (unsigned)KNN) hi = mid; else lo = mid + 1u;
    __syncthreads();
  }
  const unsigned V = lo;    // bit pattern of the 128th smallest d2

  if (l == 0) {
    gfill[g] = 0;
    pcnt[g]  = 0;
    radS[g]  = sqrtf(__uint_as_float(V));
  }
  if (l < 8) covS[g][l] = 0.0f;
  __syncthreads();

  // gather strict-less, then ties at V up to 128 total
  for (int j = l; j < NPTS; j += 64) {
    float dv = d2s[g * NPTS + j];
    if (__float_as_uint(dv) < V) {
      int pos = atomicAdd(&gfill[g], 1);
      sel_idx[g][pos] = j; sel_d2[g][pos] = dv;
    }
  }
  __syncthreads();
  for (int j = l; j < NPTS; j += 64) {
    float dv = d2s[g * NPTS + j];
    if (__float_as_uint(dv) == V) {
      int pos = atomicAdd(&gfill[g], 1);
      if (pos < KNN) { sel_idx[g][pos] = j; sel_d2[g][pos] = dv; }
    }
  }
  __syncthreads();

  // ---- weighted covariance: w = radius - d ----
  float4 qv = qpt[g];
  float radius = radS[g];
  float nxv[2], nyv[2], nzv[2];
  float acc[7] = {0, 0, 0, 0, 0, 0, 0};
#pragma unroll
  for (int kk = 0; kk < 2; ++kk) {
    int k = l + kk * 64;
    int j = sel_idx[g][k];
    float4 p = P[j];
    float x = p.x - qv.x, y = p.y - qv.y, z = p.z - qv.z;
    nxv[kk] = x; nyv[kk] = y; nzv[kk] = z;
    float dd = sqrtf(sel_d2[g][k]);
    float w  = radius - dd;
    acc[0] += w * x * x; acc[1] += w * x * y; acc[2] += w * x * z;
    acc[3] += w * y * y; acc[4] += w * y * z; acc[5] += w * z * z;
    acc[6] += w;
  }
#pragma unroll
  for (int c = 0; c < 7; ++c) atomicAdd(&covS[g][c], acc[c]);  // ds_add_f32
  __syncthreads();

  if (tid < QPB) {
    int q = tid;
    float inv = 1.0f / covS[q][6];
    smallest_evec(covS[q][0] * inv, covS[q][1] * inv, covS[q][2] * inv,
                  covS[q][3] * inv, covS[q][4] * inv, covS[q][5] * inv, &zv[q][0]);
  }
  __syncthreads();

  // ---- sign disambiguation by majority projection ----
  int pos = 0;
#pragma unroll
  for (int kk = 0; kk < 2; ++kk) {
    float zp = nxv[kk] * zv[g][0] + nyv[kk] * zv[g][1] + nzv[kk] * zv[g][2];
    pos += (zp >= 0.0f) ? 1 : 0;
  }
  atomicAdd(&pcnt[g], pos);
  __syncthreads();

  if (tid < QPB) {
    int q = tid;
    float s = (2 * pcnt[q] >= KNN) ? 1.0f : -1.0f;
    normals[(size_t)b * NPTS + qbase + q] =
        make_float4(zv[q][0] * s, zv[q][1] * s, zv[q][2] * s, 0.0f);
  }
#pragma unroll
  for (int kk = 0; kk < 2; ++kk) {
    int k = l + kk * 64;
    idxOut[((size_t)b * NPTS + qbase + g) * KNN + k] = (unsigned short)sel_idx[g][k];
  }
}

// ---------------------------------------------------------------------------
// Kernel 2: average neighbor normals and normalize
// ---------------------------------------------------------------------------
__global__ __launch_bounds__(256) void avg_normals_kernel(const float4* __restrict__ normals,
                                                          const unsigned short* __restrict__ idxOut,
                                                          float* __restrict__ out) {
  int gpt = blockIdx.x * 256 + threadIdx.x;          // 0..16383
  int b = gpt >> 13;
  const float4* nb = normals + (size_t)b * NPTS;
  const unsigned short* il = idxOut + (size_t)gpt * KNN;
  float ax = 0.0f, ay = 0.0f, az = 0.0f;
  for (int k = 0; k < KNN; ++k) {
    float4 nv = nb[il[k]];
    ax += nv.x; ay += nv.y; az += nv.z;
  }
  const float invk = 1.0f / (float)KNN;
  ax *= invk; ay *= invk; az *= invk;
  float s = rsqrtf(ax * ax + ay * ay + az * az);
  out[3 * gpt + 0] = ax * s;
  out[3 * gpt + 1] = ay * s;
  out[3 * gpt + 2] = az * s;
}

// ---------------------------------------------------------------------------
extern "C" void kernel_launch(void* const* d_in, const int* in_sizes, int n_in,
                              void* d_out, int out_size, void* d_ws, size_t ws_size,
                              hipStream_t stream) {
  (void)in_sizes; (void)n_in; (void)out_size; (void)ws_size;
  const float* verts = (const float*)d_in[0];          // (2, 8192, 3) f32
  float* out = (float*)d_out;                          // (2, 8192, 3) f32

  char* ws = (char*)d_ws;
  float4* pts4            = (float4*)(ws);                       // 16384 * 16B = 256 KB
  float4* normals         = (float4*)(ws + 262144);              // 16384 * 16B = 256 KB
  unsigned short* idxOut  = (unsigned short*)(ws + 524288);      // 16384 * 128 * 2B = 4 MB

  const int total = NSAMP * NPTS;                                // 16384
  pack_pts_kernel<<<(total + 255) / 256, 256, 0, stream>>>(verts, pts4, total);
  knn_lrf_kernel<<<NSAMP * (NPTS / QPB), 256, 0, stream>>>(pts4, normals, idxOut);
  avg_normals_kernel<<<total / 256, 256, 0, stream>>>(normals, idxOut, out);
}